// RandomShift_9380208574631
// MI455X (gfx1250) — compile-verified
//
#include <hip/hip_runtime.h>

// RandomShift: out[n,c,i,j] = bilinear 4-tap gather from x[n,c] at
//   col = i*91/83 + shift[n,0], row = j*91/83 + shift[n,1]  (zero OOB).
// HBM-bound (~260 MB total traffic). Strategy: one (n,c) plane (28 KB) per
// workgroup staged into LDS via CDNA5 async global->LDS (ASYNCcnt), gather
// from LDS, non-temporal coalesced stores.

#ifndef PADDING
#define PADDING 4
#endif

namespace {
constexpr int kN = 512;
constexpr int kC = 9;
constexpr int kH = 84;
constexpr int kW = 84;
constexpr int kPlane  = kH * kW;        // 7056 floats = 28224 B (16B aligned)
constexpr int kPlane4 = kPlane / 4;     // 1764 float4
constexpr int kHP = kH + 2 * PADDING;   // 92
constexpr int kThreads = 256;           // 8 wave32 waves
}

#if __has_builtin(__builtin_amdgcn_global_load_async_to_lds_b128)
#define ASYNC_B128 1
#elif __has_builtin(__builtin_amdgcn_global_load_async_to_lds_b32)
#define ASYNC_B32 1
#endif

// Builtin prototypes take typed pointers: (T addrspace(1)*, T addrspace(3)*, Iimm, Iimm)
typedef int v4i __attribute__((ext_vector_type(4)));
typedef v4i __attribute__((address_space(1))) *g_v4i_t;   // global int4*
typedef v4i __attribute__((address_space(3))) *l_v4i_t;   // LDS int4*
typedef int __attribute__((address_space(1))) *g_i32_t;   // global int*
typedef int __attribute__((address_space(3))) *l_i32_t;   // LDS int*

static __device__ __forceinline__ void wait_async_zero() {
#if __has_builtin(__builtin_amdgcn_s_wait_asynccnt)
    __builtin_amdgcn_s_wait_asynccnt(0);
#else
    asm volatile("s_wait_asynccnt 0" ::: "memory");
#endif
}

__global__ __launch_bounds__(kThreads)
void random_shift_kernel(const float* __restrict__ x,
                         const int*   __restrict__ shift,
                         float*       __restrict__ out)
{
    __shared__ __align__(16) float tile[kPlane];

    const int plane_id = blockIdx.x;          // n * C + c
    const int n = plane_id / kC;
    const float* __restrict__ src = x   + (size_t)plane_id * kPlane;
    float*       __restrict__ dst = out + (size_t)plane_id * kPlane;
    const int tid = threadIdx.x;

    // ---- Stage x[n,c,:,:] into LDS (coalesced 16B transfers) ----
#if defined(ASYNC_B128)
    for (int k = tid; k < kPlane4; k += kThreads) {
        __builtin_amdgcn_global_load_async_to_lds_b128(
            (g_v4i_t)(src + 4 * k), (l_v4i_t)(&tile[4 * k]), 0, 0);
    }
    wait_async_zero();
#elif defined(ASYNC_B32)
    for (int k = tid; k < kPlane; k += kThreads) {
        __builtin_amdgcn_global_load_async_to_lds_b32(
            (g_i32_t)(src + k), (l_i32_t)(&tile[k]), 0, 0);
    }
    wait_async_zero();
#else
    for (int k = tid; k < kPlane4; k += kThreads) {
        const float4 v = ((const float4*)src)[k];
        ((float4*)tile)[k] = v;
    }
#endif
    __syncthreads();

    // Per-image integer shift (column uses shift[n,0], row uses shift[n,1]).
    const float s0 = (float)shift[2 * n + 0];
    const float s1 = (float)shift[2 * n + 1];

    // Replicate reference f32 math exactly.
    const float eps   = 1.0f / (float)kHP;
    const float start = -1.0f + eps;
    const float stop  =  1.0f - eps;
    const float step  = (stop - start) / (float)(kH - 1);
    const float sc    = 2.0f / (float)kHP;

    for (int idx = tid; idx < kPlane; idx += kThreads) {
        const int i = idx / kW;
        const int j = idx - i * kW;

        const float gx = (start + (float)i * step) + s0 * sc;  // grid[...,0] -> x (col)
        const float gy = (start + (float)j * step) + s1 * sc;  // grid[...,1] -> y (row)

        const float ixf = ((gx + 1.0f) * (float)kHP - 1.0f) * 0.5f;
        const float iyf = ((gy + 1.0f) * (float)kHP - 1.0f) * 0.5f;

        const float fx0 = floorf(ixf);
        const float fy0 = floorf(iyf);
        const float wx1 = ixf - fx0, wx0 = 1.0f - wx1;
        const float wy1 = iyf - fy0, wy0 = 1.0f - wy1;

        // Indices into the UNpadded plane (padding ring is zeros anyway,
        // and reference zero-masks out-of-range taps).
        const int c0 = (int)fx0 - PADDING;
        const int c1 = c0 + 1;
        const int r0 = (int)fy0 - PADDING;
        const int r1 = r0 + 1;

        const bool cv0 = (unsigned)c0 < (unsigned)kW;
        const bool cv1 = (unsigned)c1 < (unsigned)kW;
        const bool rv0 = (unsigned)r0 < (unsigned)kH;
        const bool rv1 = (unsigned)r1 < (unsigned)kH;

        const float v00 = (rv0 && cv0) ? tile[r0 * kW + c0] : 0.0f;
        const float v01 = (rv0 && cv1) ? tile[r0 * kW + c1] : 0.0f;
        const float v10 = (rv1 && cv0) ? tile[r1 * kW + c0] : 0.0f;
        const float v11 = (rv1 && cv1) ? tile[r1 * kW + c1] : 0.0f;

        // Same tap order / association as reference: ((t00+t01)+t10)+t11.
        const float t00 = v00 * (wy0 * wx0);
        const float t01 = v01 * (wy0 * wx1);
        const float t10 = v10 * (wy1 * wx0);
        const float t11 = v11 * (wy1 * wx1);
        const float acc = ((t00 + t01) + t10) + t11;

        __builtin_nontemporal_store(acc, &dst[idx]);  // output never re-read
    }
}

extern "C" void kernel_launch(void* const* d_in, const int* in_sizes, int n_in,
                              void* d_out, int out_size, void* d_ws, size_t ws_size,
                              hipStream_t stream) {
    const float* x     = (const float*)d_in[0];   // [512,9,84,84] f32
    const int*   shift = (const int*)d_in[1];     // [512,2] i32
    float*       out   = (float*)d_out;           // [512,9,84,84] f32
    (void)in_sizes; (void)n_in; (void)out_size; (void)d_ws; (void)ws_size;

    random_shift_kernel<<<dim3(kN * kC), dim3(kThreads), 0, stream>>>(x, shift, out);
}